// ContactLoss_61830349193761
// MI455X (gfx1250) — compile-verified
//
#include <hip/hip_runtime.h>
#include <math.h>

typedef __attribute__((ext_vector_type(2))) float v2f;
typedef __attribute__((ext_vector_type(8))) float v8f;

#define B_   16
#define VS_  10475
#define VO_  8192
#define K_   64
#define PS_  1024
#define PO_  2048

#define STILE  128   // SMPL points per workgroup
#define BLOCK  256   // 8 waves (wave32)

__global__ void contact_zero_kernel(float* out) {
    if (threadIdx.x == 0 && blockIdx.x == 0) out[0] = 0.0f;
}

__launch_bounds__(BLOCK)
__global__ void ContactLoss_61830349193761_kernel(
        const float* __restrict__ smplx_v,      // (B, VS, 3)
        const float* __restrict__ object_v,     // (B, VO, 3)
        const int*   __restrict__ smpl_part_idx,// (K, PS)
        const int*   __restrict__ obj_part_idx, // (K, PO)
        const int*   __restrict__ batch_idx,    // (K,)
        float*       __restrict__ out)
{
    // Packed 16B records {x,y,z,0}: one unconditional ds_load_b64 yields the
    // (K0,K1)=(x,y) pair for lanes 0-15 or (K2,K3)=(z,0) pair for lanes 16-31.
    __shared__ float spack[STILE][4];
    __shared__ float sn[STILE];
    __shared__ float opack[PO_][4];
    __shared__ float on[PO_];
    __shared__ float wg_sum;

    const int k     = blockIdx.x;   // part index 0..63
    const int stile = blockIdx.y;   // S tile 0..7
    const int tid   = threadIdx.x;
    const int b     = batch_idx[k]; // scalar broadcast

    if (tid == 0) wg_sum = 0.0f;

    // ---- stage the 2048 gathered object points + squared norms into LDS ----
    for (int i = tid; i < PO_; i += BLOCK) {
        const int oi = obj_part_idx[k * PO_ + i];
        const float* p = object_v + ((size_t)b * VO_ + (size_t)oi) * 3;
        const float x = p[0], y = p[1], z = p[2];
        opack[i][0] = x; opack[i][1] = y; opack[i][2] = z; opack[i][3] = 0.0f;
        on[i] = x * x + y * y + z * z;
    }
    // ---- stage this workgroup's 128 gathered SMPL points ----
    if (tid < STILE) {
        const int si = smpl_part_idx[k * PS_ + stile * STILE + tid];
        const float* p = smplx_v + ((size_t)b * VS_ + (size_t)si) * 3;
        const float x = p[0], y = p[1], z = p[2];
        spack[tid][0] = x; spack[tid][1] = y; spack[tid][2] = z; spack[tid][3] = 0.0f;
        sn[tid] = x * x + y * y + z * z;
    }
    __syncthreads();

    const int  wave  = tid >> 5;
    const int  lane  = tid & 31;
    const int  l     = lane & 15;
    const bool hi    = lane >= 16;
    const int  hioff = hi ? 2 : 0;      // float offset into the 16B record
    const int  sBase = wave * 16;       // 16 S rows per wave

    // A fragment (16x4 f32, 2 VGPRs): one b64 load per lane, no divergence
    const v2f afrag = *reinterpret_cast<const v2f*>(&spack[sBase + l][hioff]);

    // ||s||^2 for each accumulator row r: C element (r,lane) is M = r + (hi?8:0)
    float nsr[8];
    #pragma unroll
    for (int r = 0; r < 8; ++r)
        nsr[r] = sn[sBase + r + (hi ? 8 : 0)];

    // running min of (||o||^2 - 2 s.o) per C row (sqrt & ||s||^2 deferred:
    // min is monotone, so clamp+sqrt happen once after the O sweep)
    float mn[8];
    #pragma unroll
    for (int r = 0; r < 8; ++r) mn[r] = 3.0e38f;

    #pragma unroll 4
    for (int j = 0; j < PO_; j += 16) {
        const int n = j + l;
        // B fragment (4x16 f32): single unconditional ds_load_b64
        const v2f bfrag = *reinterpret_cast<const v2f*>(&opack[n][hioff]);

        v8f acc = {0.f, 0.f, 0.f, 0.f, 0.f, 0.f, 0.f, 0.f};
        // D = A(16x4) x B(4x16) + 0  -> per-element s.o dot products
        acc = __builtin_amdgcn_wmma_f32_16x16x4_f32(
                  /*neg_a=*/false, afrag, /*neg_b=*/false, bfrag,
                  /*c_mod=*/(short)0, acc, /*reuse_a=*/false, /*reuse_b=*/false);

        const float nov = on[n];    // ||o_N||^2, N = j + (lane&15) in both halves
        #pragma unroll
        for (int r = 0; r < 8; ++r) {
            const float t = __builtin_fmaf(-2.0f, acc[r], nov); // no - 2*dot
            mn[r] = fminf(mn[r], t);
        }
    }

    // min-reduce over N: xor-shuffle within each 16-lane half of the wave32
    #pragma unroll
    for (int r = 0; r < 8; ++r) {
        float v = mn[r];
        #pragma unroll
        for (int m = 8; m >= 1; m >>= 1)
            v = fminf(v, __shfl_xor(v, m, 32));
        mn[r] = v;
    }

    // lanes 0 and 16 hold the 16 per-S-point minima for this wave
    if (l == 0) {
        float s = 0.0f;
        #pragma unroll
        for (int r = 0; r < 8; ++r) {
            const float d2 = fmaxf(nsr[r] + mn[r], 0.0f);
            s += __builtin_sqrtf(d2);
        }
        atomicAdd(&wg_sum, s);     // ds_add_f32
    }
    __syncthreads();

    if (tid == 0)
        atomicAdd(out, wg_sum * (1.0f / (float)(K_ * PS_)));
}

extern "C" void kernel_launch(void* const* d_in, const int* in_sizes, int n_in,
                              void* d_out, int out_size, void* d_ws, size_t ws_size,
                              hipStream_t stream) {
    (void)in_sizes; (void)n_in; (void)out_size; (void)d_ws; (void)ws_size;
    const float* smplx_v       = (const float*)d_in[0];
    const float* object_v      = (const float*)d_in[1];
    const int*   smpl_part_idx = (const int*)  d_in[2];
    const int*   obj_part_idx  = (const int*)  d_in[3];
    const int*   batch_idx     = (const int*)  d_in[4];
    float* out = (float*)d_out;

    contact_zero_kernel<<<1, 32, 0, stream>>>(out);

    dim3 grid(K_, PS_ / STILE);   // 64 x 8 = 512 workgroups
    ContactLoss_61830349193761_kernel<<<grid, BLOCK, 0, stream>>>(
        smplx_v, object_v, smpl_part_idx, obj_part_idx, batch_idx, out);
}